// Demosaic_50208167690678
// MI455X (gfx1250) — compile-verified
//
#include <hip/hip_runtime.h>
#include <cstdint>
#include <cstddef>

// ---------------------------------------------------------------------------
// Malvar-He-Cutler demosaic, RGGB, fp32. Memory-bound streaming stencil:
//   traffic floor ~268 MB -> ~11.5us @ 23.3 TB/s; FLOPs ~0.7 GFLOP (noise).
// Strategy: async global->LDS staging (gfx1250 ASYNCcnt path), haloed 64x16
// tiles, one 2x2 bayer quad per thread (branch-free select), float2
// nontemporal coalesced stores.
// ---------------------------------------------------------------------------

typedef float fvec2 __attribute__((ext_vector_type(2)));

#define AS1 __attribute__((address_space(1)))
#define AS3 __attribute__((address_space(3)))

#ifndef __has_builtin
#define __has_builtin(x) 0
#endif

__device__ __forceinline__ void async_load_b32(float* lds_dst, const float* g_src) {
#if __has_builtin(__builtin_amdgcn_global_load_async_to_lds_b32)
    __builtin_amdgcn_global_load_async_to_lds_b32(
        (AS1 int*)(uintptr_t)g_src,
        (AS3 int*)(uint32_t)(uintptr_t)lds_dst,   // low 32 bits of generic ptr == LDS byte addr
        0, 0);
#else
    uint32_t l = (uint32_t)(uintptr_t)lds_dst;
    uint64_t g = (uint64_t)(uintptr_t)g_src;
    asm volatile("global_load_async_to_lds_b32 %0, %1, off" :: "v"(l), "v"(g) : "memory");
#endif
}

__device__ __forceinline__ void wait_async0() {
#if __has_builtin(__builtin_amdgcn_s_wait_asynccnt)
    __builtin_amdgcn_s_wait_asynccnt(0);
#else
    asm volatile("s_wait_asynccnt 0x0" ::: "memory");
#endif
}

constexpr int H  = 1024;
constexpr int W  = 1024;
constexpr int TW = 64;        // tile width  (pixels)
constexpr int THt = 16;       // tile height (pixels)
constexpr int LW = TW + 4;    // 68: LDS row length incl. 2-halo
constexpr int LH = THt + 4;   // 20
constexpr int LDS_ELEMS = LW * LH;  // 1360 floats = 5440 B

__global__ __launch_bounds__(256, 4)
void demosaic_mhc_rggb(const float* __restrict__ x, float* __restrict__ out, int Hq /*unused pad*/) {
    __shared__ float tile[LDS_ELEMS];

    const int t   = threadIdx.x;
    const int tx0 = blockIdx.x * TW;
    const int ty0 = blockIdx.y * THt;
    const int n   = blockIdx.z;

    const float* __restrict__ src = x + (size_t)n * H * W;   // C_in == 1

    // ---- Stage haloed input tile into LDS via async copy engine -----------
    for (int i = t; i < LDS_ELEMS; i += 256) {
        const int r = i / LW;
        const int c = i - r * LW;
        int gy = ty0 - 2 + r; gy = gy < 0 ? 0 : (gy > H - 1 ? H - 1 : gy);  // edge replicate
        int gx = tx0 - 2 + c; gx = gx < 0 ? 0 : (gx > W - 1 ? W - 1 : gx);
        async_load_b32(&tile[i], src + (size_t)gy * W + gx);
    }
    wait_async0();
    __syncthreads();

    // ---- One 2x2 bayer quad per thread ------------------------------------
    const int qx = t & 31;         // lanes 0..31 -> 64 contiguous pixels / wave
    const int qy = t >> 5;         // 8 quad rows -> 16 pixel rows
    const int bx = qx * 2;         // LDS col of 6x6 window origin
    const int by = qy * 2;         // LDS row of 6x6 window origin

    float w[6][6];
#pragma unroll
    for (int r = 0; r < 6; ++r) {
        const fvec2* p = (const fvec2*)&tile[(by + r) * LW + bx];   // 8B aligned, bank-conflict-free
        fvec2 a = p[0], b = p[1], c = p[2];
        w[r][0] = a.x; w[r][1] = a.y; w[r][2] = b.x;
        w[r][3] = b.y; w[r][4] = c.x; w[r][5] = c.y;
    }

    // MHC kernels (x16), applied at window offset (py,px), center w[py+2][px+2]
    auto C0 = [&](int py, int px) {   // cross: G at R/B sites
        return 8.0f * w[py+2][px+2]
             + 4.0f * (w[py+1][px+2] + w[py+3][px+2] + w[py+2][px+1] + w[py+2][px+3])
             - 2.0f * (w[py  ][px+2] + w[py+4][px+2] + w[py+2][px  ] + w[py+2][px+4]);
    };
    auto C1 = [&](int py, int px) {   // checker: B at R site / R at B site
        return 12.0f * w[py+2][px+2]
             + 4.0f * (w[py+1][px+1] + w[py+1][px+3] + w[py+3][px+1] + w[py+3][px+3])
             - 3.0f * (w[py  ][px+2] + w[py+4][px+2] + w[py+2][px  ] + w[py+2][px+4]);
    };
    auto C2 = [&](int py, int px) {   // horizontal
        return 10.0f * w[py+2][px+2]
             + 8.0f * (w[py+2][px+1] + w[py+2][px+3])
             - 2.0f * (w[py+1][px+1] + w[py+1][px+3] + w[py+3][px+1] + w[py+3][px+3]
                     + w[py+2][px  ] + w[py+2][px+4])
             + 1.0f * (w[py  ][px+2] + w[py+4][px+2]);
    };
    auto C3 = [&](int py, int px) {   // vertical
        return 10.0f * w[py+2][px+2]
             + 8.0f * (w[py+1][px+2] + w[py+3][px+2])
             - 2.0f * (w[py+1][px+1] + w[py+1][px+3] + w[py+3][px+1] + w[py+3][px+3]
                     + w[py  ][px+2] + w[py+4][px+2])
             + 1.0f * (w[py+2][px  ] + w[py+2][px+4]);
    };

    const float s = 1.0f / 16.0f;
    // Quadrant (0,0): R site   -> R=raw, G=k0, B=k1
    const float R00 = w[2][2];
    const float G00 = C0(0, 0) * s;
    const float B00 = C1(0, 0) * s;
    // Quadrant (0,1): G site   -> R=k2, G=raw, B=k3
    const float R01 = C2(0, 1) * s;
    const float G01 = w[2][3];
    const float B01 = C3(0, 1) * s;
    // Quadrant (1,0): G site   -> R=k3, G=raw, B=k2
    const float R10 = C3(1, 0) * s;
    const float G10 = w[3][2];
    const float B10 = C2(1, 0) * s;
    // Quadrant (1,1): B site   -> R=k1, G=k0, B=raw
    const float R11 = C1(1, 1) * s;
    const float G11 = C0(1, 1) * s;
    const float B11 = w[3][3];

    // ---- Coalesced nontemporal float2 stores (write-once output) ----------
    const int oy = ty0 + by;
    const int ox = tx0 + bx;
    const size_t base = (size_t)n * 3 * H * W;
    auto store2 = [&](int ch, int row, float v0, float v1) {
        fvec2 v; v.x = v0; v.y = v1;
        fvec2* p = (fvec2*)(out + base + (size_t)ch * H * W + (size_t)(oy + row) * W + ox);
        __builtin_nontemporal_store(v, p);
    };
    store2(0, 0, R00, R01); store2(0, 1, R10, R11);
    store2(1, 0, G00, G01); store2(1, 1, G10, G11);
    store2(2, 0, B00, B01); store2(2, 1, B10, B11);
}

extern "C" void kernel_launch(void* const* d_in, const int* in_sizes, int n_in,
                              void* d_out, int out_size, void* d_ws, size_t ws_size,
                              hipStream_t stream) {
    const float* x  = (const float*)d_in[0];
    // d_in[1] = bayer_pattern: static metadata (RGGB) in the reference; hardcoded.
    float* out = (float*)d_out;

    const int N = in_sizes[0] / (H * W);   // 16
    dim3 grid(W / TW, H / THt, N);         // (16, 64, 16)
    demosaic_mhc_rggb<<<grid, 256, 0, stream>>>(x, out, 0);
}